// DSNTDistanceAngleDoubleLoss_old_20083267076472
// MI455X (gfx1250) — compile-verified
//
#include <hip/hip_runtime.h>
#include <math.h>
#include <stdint.h>

// DSNT distance/angle loss: memory-bound streaming reduction.
// 256 MB total traffic, ~11us floor at 23.3 TB/s. Strategy: read every byte
// exactly once; stage input chunks into LDS via CDNA5 async global->LDS ops
// (ASYNCcnt path), two-phase exact softmax per chunk (block-local max), fold
// the 8 per-wave partials with V_WMMA_F32_16X16X4_F32 (exact f32 matrix op),
// merge partials with log-sum-exp scaling in a tiny finalize kernel.

#define NB 64
#define NP 2
#define NMAPS (NB * NP)      // 128 heatmaps
#define HDIM 512
#define WDIM 512
#define HW (HDIM * WDIM)     // 262144 elements per heatmap
#define BPM 32               // chunks per heatmap
#define CHUNK (HW / BPM)     // 8192 elements = 32 KB staged in LDS
#define BDIM 256             // 8 waves (wave32)
#define CH4 (CHUNK / 4)      // 2048 float4 per chunk
#define ITER (CH4 / BDIM)    // 8 float4 per thread
#define WSSTRIDE 6           // per-block partial record (floats)

#define AS1 __attribute__((address_space(1)))
#define AS3 __attribute__((address_space(3)))

typedef int   v4i __attribute__((ext_vector_type(4)));
typedef float v2f __attribute__((ext_vector_type(2)));
typedef float v8f __attribute__((ext_vector_type(8)));

#if defined(__HIP_DEVICE_COMPILE__) && __has_builtin(__builtin_amdgcn_global_load_async_to_lds_b128)
#define ASYNC_LDS 1
#else
#define ASYNC_LDS 0
#endif

#if defined(__HIP_DEVICE_COMPILE__) && __has_builtin(__builtin_amdgcn_wmma_f32_16x16x4_f32)
#define HAVE_WMMA 1
#else
#define HAVE_WMMA 0
#endif

__device__ __forceinline__ float wave_max_f(float v) {
#pragma unroll
  for (int m = 16; m >= 1; m >>= 1) v = fmaxf(v, __shfl_xor(v, m, 32));
  return v;
}

__device__ __forceinline__ float wave_sum_f(float v) {
#pragma unroll
  for (int m = 16; m >= 1; m >>= 1) v += __shfl_xor(v, m, 32);
  return v;
}

__device__ __forceinline__ void wave_argmax(float& v, int& i) {
#pragma unroll
  for (int m = 16; m >= 1; m >>= 1) {
    float v2 = __shfl_xor(v, m, 32);
    int   i2 = __shfl_xor(i, m, 32);
    if (v2 > v || (v2 == v && i2 < i)) { v = v2; i = i2; }
  }
}

__global__ void __launch_bounds__(BDIM)
dsnt_partial_kernel(const float* __restrict__ inp, const float* __restrict__ tgt,
                    float* __restrict__ ws) {
  __shared__ float4 sm4[CH4];            // 32 KB staging tile
  __shared__ float  red3[3 * 8];         // [quantity q][wave w] -> red3[q*8+w]
  __shared__ float  redM[8];
  __shared__ float  redTV[8];
  __shared__ int    redTI[8];
  __shared__ float  bcastM;

  const int tid   = threadIdx.x;
  const int wave  = tid >> 5;
  const int lane  = tid & 31;
  const int blk   = blockIdx.x;
  const int map   = blk / BPM;
  const int chunk = blk % BPM;
  const size_t base4 = (size_t)map * (HW / 4) + (size_t)chunk * CH4;

  const float4* in4 = (const float4*)inp;
  const float4* tg4 = (const float4*)tgt;

  // ---- stage input chunk into LDS (async on CDNA5) ----
#if ASYNC_LDS
#pragma unroll
  for (int i = 0; i < ITER; ++i) {
    const int idx = i * BDIM + tid;
    __builtin_amdgcn_global_load_async_to_lds_b128(
        (AS1 v4i*)(uintptr_t)(in4 + base4 + idx),
        (AS3 v4i*)(uintptr_t)(sm4 + idx),
        0, 0);
  }
#else
#pragma unroll
  for (int i = 0; i < ITER; ++i) {
    const int idx = i * BDIM + tid;
    sm4[idx] = in4[base4 + idx];
  }
#endif

  // ---- overlap: target max/argmax from global (first-index tie-break) ----
  float tv = -INFINITY;
  int   ti = 0;
#pragma unroll
  for (int i = 0; i < ITER; ++i) {
    const int idx = i * BDIM + tid;
    const float4 t = tg4[base4 + idx];
    const int fi = chunk * CHUNK + idx * 4;  // flat index within heatmap
    if (t.x > tv) { tv = t.x; ti = fi; }
    if (t.y > tv) { tv = t.y; ti = fi + 1; }
    if (t.z > tv) { tv = t.z; ti = fi + 2; }
    if (t.w > tv) { tv = t.w; ti = fi + 3; }
  }
  wave_argmax(tv, ti);
  if (lane == 0) { redTV[wave] = tv; redTI[wave] = ti; }

  // ---- wait for async staging, then make LDS visible block-wide ----
#if ASYNC_LDS
#if __has_builtin(__builtin_amdgcn_s_wait_asynccnt)
  __builtin_amdgcn_s_wait_asynccnt(0);
#else
  asm volatile("s_wait_asynccnt 0" ::: "memory");
#endif
#endif
  __syncthreads();

  if (tid == 0) {  // fold 8 wave argmax partials
    float v = redTV[0]; int ix = redTI[0];
#pragma unroll
    for (int k = 1; k < 8; ++k)
      if (redTV[k] > v || (redTV[k] == v && redTI[k] < ix)) { v = redTV[k]; ix = redTI[k]; }
    redTV[0] = v; redTI[0] = ix;
  }

  // ---- phase A: block-local max of input chunk (exact softmax shift) ----
  float m = -INFINITY;
#pragma unroll
  for (int i = 0; i < ITER; ++i) {
    const float4 v = sm4[i * BDIM + tid];
    m = fmaxf(m, fmaxf(fmaxf(v.x, v.y), fmaxf(v.z, v.w)));
  }
  m = wave_max_f(m);
  if (lane == 0) redM[wave] = m;
  __syncthreads();
  if (tid == 0) {
    float mm = redM[0];
#pragma unroll
    for (int k = 1; k < 8; ++k) mm = fmaxf(mm, redM[k]);
    bcastM = mm;
  }
  __syncthreads();
  const float bm = bcastM;

  // ---- phase B: exp sums + coordinate-weighted sums ----
  float s = 0.f, sx = 0.f, sy = 0.f;
#pragma unroll
  for (int i = 0; i < ITER; ++i) {
    const int idx = i * BDIM + tid;
    const float4 v = sm4[idx];
    const int fi = chunk * CHUNK + idx * 4;
    const float row1 = (float)((fi >> 9) + 1);      // (row + 1), same for 4 lanes
    const float c1   = (float)((fi & 511) + 1);     // (col + 1) of .x
    const float e0 = __expf(v.x - bm);
    const float e1 = __expf(v.y - bm);
    const float e2 = __expf(v.z - bm);
    const float e3 = __expf(v.w - bm);
    const float es = e0 + e1 + e2 + e3;
    s  += es;
    sy += es * row1;
    sx += e0 * c1 + e1 * (c1 + 1.f) + e2 * (c1 + 2.f) + e3 * (c1 + 3.f);
  }
  s  = wave_sum_f(s);
  sx = wave_sum_f(sx);
  sy = wave_sum_f(sy);
  if (lane == 0) {
    red3[0 * 8 + wave] = s;
    red3[1 * 8 + wave] = sx;
    red3[2 * 8 + wave] = sy;
  }
  __syncthreads();

  // ---- fold 8 wave partials: D = A x ones via V_WMMA_F32_16X16X4_F32 ----
  // A[m][k]: row m = quantity (0:S 1:SX 2:SY), K slot k = wave index.
  // A layout (16x4 f32): lanes 0-15 hold K=0,1 in vgpr0,1; lanes 16-31 hold K=2,3.
  // Two chained WMMAs (C-accumulate) cover waves 0..3 and 4..7. B == all-ones so
  // its layout is irrelevant. D layout: lane 0 vgpr j = D[j][0] -> S,SX,SY.
  // Executed by every wave with full EXEC (no divergence; WMMA requires EXEC all-1).
#if HAVE_WMMA
  {
    const int mrow  = lane & 15;
    const int kbase = (lane >> 4) << 1;  // 0 or 2
    v2f a1 = {0.f, 0.f}, a2 = {0.f, 0.f};
    if (mrow < 3) {
      a1.x = red3[mrow * 8 + kbase + 0];
      a1.y = red3[mrow * 8 + kbase + 1];
      a2.x = red3[mrow * 8 + 4 + kbase + 0];
      a2.y = red3[mrow * 8 + 4 + kbase + 1];
    }
    const v2f bones = {1.f, 1.f};
    v8f d = {};
    d = __builtin_amdgcn_wmma_f32_16x16x4_f32(false, a1, false, bones, (short)0, d, false, false);
    d = __builtin_amdgcn_wmma_f32_16x16x4_f32(false, a2, false, bones, (short)0, d, false, false);
    if (tid == 0) {
      float* p = ws + (size_t)blk * WSSTRIDE;
      p[0] = bm;
      p[1] = d[0];
      p[2] = d[1];
      p[3] = d[2];
      p[4] = redTV[0];
      p[5] = __int_as_float(redTI[0]);
    }
  }
#else
  if (tid == 0) {
    float S = 0.f, SX = 0.f, SY = 0.f;
#pragma unroll
    for (int k = 0; k < 8; ++k) {
      S  += red3[0 * 8 + k];
      SX += red3[1 * 8 + k];
      SY += red3[2 * 8 + k];
    }
    float* p = ws + (size_t)blk * WSSTRIDE;
    p[0] = bm;
    p[1] = S;
    p[2] = SX;
    p[3] = SY;
    p[4] = redTV[0];
    p[5] = __int_as_float(redTI[0]);
  }
#endif
}

__global__ void __launch_bounds__(NMAPS)
dsnt_finalize_kernel(const float* __restrict__ ws, float* __restrict__ out) {
  __shared__ float px[NMAPS], py[NMAPS], txs[NMAPS], tys[NMAPS];
  __shared__ float lsum[NB];
  const int t = threadIdx.x;

  if (t < NMAPS) {
    const float* p = ws + (size_t)t * BPM * WSSTRIDE;
    float M = -INFINITY;
#pragma unroll 4
    for (int k = 0; k < BPM; ++k) M = fmaxf(M, p[k * WSSTRIDE + 0]);
    float S = 0.f, SX = 0.f, SY = 0.f;
    float tv = -INFINITY; int ti = 0;
#pragma unroll 4
    for (int k = 0; k < BPM; ++k) {
      const float sc = __expf(p[k * WSSTRIDE + 0] - M);  // log-sum-exp merge
      S  += p[k * WSSTRIDE + 1] * sc;
      SX += p[k * WSSTRIDE + 2] * sc;
      SY += p[k * WSSTRIDE + 3] * sc;
      const float v = p[k * WSSTRIDE + 4];
      const int  ix = __float_as_int(p[k * WSSTRIDE + 5]);
      if (v > tv || (v == tv && ix < ti)) { tv = v; ti = ix; }
    }
    px[t]  = SX / (S * (float)WDIM);
    py[t]  = SY / (S * (float)HDIM);
    txs[t] = (float)((ti % WDIM) + 1) / (float)WDIM;
    tys[t] = (float)((ti / WDIM) + 1) / (float)HDIM;
  }
  __syncthreads();

  if (t < NB) {
    const int i0 = t * 2, i1 = t * 2 + 1;
    const float dx0 = txs[i0] - px[i0], dy0 = tys[i0] - py[i0];
    const float dx1 = txs[i1] - px[i1], dy1 = tys[i1] - py[i1];
    const float ed = sqrtf(dx0 * dx0 + dy0 * dy0) + sqrtf(dx1 * dx1 + dy1 * dy1);
    const float pvx = px[i0] - px[i1], pvy = py[i0] - py[i1];
    const float tvx = txs[i0] - txs[i1], tvy = tys[i0] - tys[i1];
    const float pd = sqrtf(pvx * pvx + pvy * pvy);
    const float td = sqrtf(tvx * tvx + tvy * tvy);
    const float dot = pvx * tvx + pvy * tvy;
    const float cd = 1.0f - cosf(dot / (pd * td));  // faithful: cos of cos-sim
    lsum[t] = ed + fabsf(pd - td) + cd;
  }
  __syncthreads();

  if (t == 0) {
    float acc = 0.f;
#pragma unroll 4
    for (int k = 0; k < NB; ++k) acc += lsum[k];
    out[0] = acc / (float)NB;
  }
}

extern "C" void kernel_launch(void* const* d_in, const int* in_sizes, int n_in,
                              void* d_out, int out_size, void* d_ws, size_t ws_size,
                              hipStream_t stream) {
  const float* inp = (const float*)d_in[0];
  const float* tgt = (const float*)d_in[1];
  float* ws  = (float*)d_ws;    // NMAPS*BPM*6 floats = 96 KB of partials
  float* out = (float*)d_out;

  dsnt_partial_kernel<<<NMAPS * BPM, BDIM, 0, stream>>>(inp, tgt, ws);
  dsnt_finalize_kernel<<<1, NMAPS, 0, stream>>>(ws, out);
}